// HungarianMatcher_18485539242566
// MI455X (gfx1250) — compile-verified
//
#include <hip/hip_runtime.h>

// ---------------------------------------------------------------------------
// HungarianMatcher cost matrix for MI455X (gfx1250, wave32).
//
//   prep_cc:    cc[n,k] = COST_CLASS * (pos - neg) focal class cost  [N,16]
//   prep_boxes: CCW corners [*,4,2] + area for preds and targets
//   cost_tiles: one wave32 per 16x16 (pred x tgt) tile.
//     - each lane computes 8 pairs: 5*(L1 + angle) - 2*rotatedIoU directly
//       into the V_WMMA C-operand register layout (ISA 7.12.2); the pair loop
//       is kept ROLLED to hold the Sutherland-Hodgman code to one copy
//       (I-cache friendly: prior fully-unrolled build was ~16.7K instrs)
//     - class cost added as cc[N,16] x onehot(label)[16,16] via a chain of
//       four V_WMMA_F32_16X16X4_F32 (K=16) -> bit-exact f32 gather+add
//     - D tile stored straight to out[N,T]
// ---------------------------------------------------------------------------

typedef float v2f __attribute__((ext_vector_type(2)));
typedef float v8f __attribute__((ext_vector_type(8)));

#define EPSF      1e-6f
#define PI_F      3.14159265358979323846f
#define HALF_PI_F 1.57079632679489661923f

// Intersection area of two convex CCW quads via Sutherland-Hodgman clipping.
__device__ __forceinline__ float quadInterArea(const float* px, const float* py,
                                               const float* qx, const float* qy) {
  float X[10], Y[10];
  int n = 4;
#pragma unroll
  for (int i = 0; i < 4; ++i) { X[i] = px[i]; Y[i] = py[i]; }

#pragma unroll
  for (int e = 0; e < 4; ++e) {
    const float ax = qx[e], ay = qy[e];
    const int e1 = (e + 1) & 3;
    const float ex = qx[e1] - ax, ey = qy[e1] - ay;

    float Xo[10], Yo[10];
    int m = 0;
    float cxP = X[n - 1], cyP = Y[n - 1];
    float sP = ex * (cyP - ay) - ey * (cxP - ax);
    for (int i = 0; i < n; ++i) {
      const float cx = X[i], cy = Y[i];
      const float s = ex * (cy - ay) - ey * (cx - ax);
      const bool inC = (s >= -EPSF);
      const bool inP = (sP >= -EPSF);
      if (inC != inP) {
        const float t = sP / (sP - s);
        Xo[m] = cxP + t * (cx - cxP);
        Yo[m] = cyP + t * (cy - cyP);
        ++m;
      }
      if (inC) { Xo[m] = cx; Yo[m] = cy; ++m; }
      sP = s; cxP = cx; cyP = cy;
    }
    n = m;
    if (n == 0) return 0.0f;
    for (int i = 0; i < n; ++i) { X[i] = Xo[i]; Y[i] = Yo[i]; }
  }
  if (n < 3) return 0.0f;
  float a2 = 0.0f;
  for (int i = 0; i < n; ++i) {
    const int j = (i + 1 == n) ? 0 : i + 1;
    a2 += X[i] * Y[j] - X[j] * Y[i];
  }
  return 0.5f * fabsf(a2);
}

// cc[n,k] = COST_CLASS * (pos - neg), focal-style, ALPHA=0.25 GAMMA=2
__global__ void hm_prep_cc(const float* __restrict__ logits,
                           float* __restrict__ ccS, int total) {
  const int idx = blockIdx.x * blockDim.x + threadIdx.x;
  if (idx >= total) return;
  const float x  = logits[idx];
  const float op = 1.0f / (1.0f + expf(-x));
  const float pos = 0.25f * (1.0f - op) * (1.0f - op) * (-logf(op + 1e-8f));
  const float neg = 0.75f * op * op * (-logf(1.0f - op + 1e-8f));
  ccS[idx] = 2.0f * (pos - neg);   // COST_CLASS = 2
}

// Corners (CCW, matches reference dx/dy order) + area for all boxes.
__global__ void hm_prep_boxes(const float* __restrict__ predBoxes,
                              const float* __restrict__ tgtBoxes,
                              float* __restrict__ predCorn, float* __restrict__ predArea,
                              float* __restrict__ tgtCorn,  float* __restrict__ tgtArea,
                              int N, int T) {
  const int idx = blockIdx.x * blockDim.x + threadIdx.x;
  if (idx >= N + T) return;
  const float* box;
  float* corn;
  float* areaOut;
  if (idx < N) { box = predBoxes + (size_t)idx * 5; corn = predCorn + (size_t)idx * 8; areaOut = predArea + idx; }
  else { const int r = idx - N; box = tgtBoxes + (size_t)r * 5; corn = tgtCorn + (size_t)r * 8; areaOut = tgtArea + r; }

  const float cx = box[0], cy = box[1], w = box[2], h = box[3], a = box[4];
  const float cA = cosf(a), sA = sinf(a);
  const float dx[4] = { 0.5f, -0.5f, -0.5f,  0.5f};
  const float dy[4] = { 0.5f,  0.5f, -0.5f, -0.5f};
#pragma unroll
  for (int i = 0; i < 4; ++i) {
    const float lx = dx[i] * w, ly = dy[i] * h;
    corn[2 * i + 0] = cx + lx * cA - ly * sA;
    corn[2 * i + 1] = cy + lx * sA + ly * cA;
  }
  *areaOut = w * h;
}

// One wave32 per 16x16 tile of the [N,T] cost matrix.
__global__ __launch_bounds__(128) void hm_cost_tiles(
    const float* __restrict__ predBoxes, const float* __restrict__ tgtBoxes,
    const int*   __restrict__ tgtLabels, const float* __restrict__ ccS,
    const float* __restrict__ predCorn,  const float* __restrict__ predArea,
    const float* __restrict__ tgtCorn,   const float* __restrict__ tgtArea,
    float* __restrict__ out, int N, int T, int C, int tilesT) {
  const int wavesPerBlk = blockDim.x >> 5;
  const int w = blockIdx.x * wavesPerBlk + (threadIdx.x >> 5);
  const int tilesM = (N + 15) >> 4;
  if (w >= tilesM * tilesT) return;          // wave-uniform exit: EXEC stays all-1

  const int tm = w / tilesT, tt = w - tm * tilesT;
  const int m0 = tm << 4, t0 = tt << 4;

  const int lane = threadIdx.x & 31;
  const int half = lane >> 4;                // lane-half selects row group / K group
  const int col  = lane & 15;                // target column within tile

  int t = t0 + col; if (t >= T) t = T - 1;   // clamp loads; stores guarded below

  float tb[5];
#pragma unroll
  for (int j = 0; j < 5; ++j) tb[j] = tgtBoxes[(size_t)t * 5 + j];
  float qx[4], qy[4];
  {
    const float4* tc = (const float4*)(tgtCorn + (size_t)t * 8);
    const float4 c0 = tc[0], c1 = tc[1];
    qx[0] = c0.x; qy[0] = c0.y; qx[1] = c0.z; qy[1] = c0.w;
    qx[2] = c1.x; qy[2] = c1.y; qx[3] = c1.z; qy[3] = c1.w;
  }
  const float ta = tgtArea[t];

  // C-operand layout (ISA 7.12.2): VGPR i <-> row m0 + 8*half + i, col = lane&15.
  // Keep this loop ROLLED: one copy of the clipper (code size >> unroll benefit).
  v8f c;
#pragma clang loop unroll(disable)
  for (int i = 0; i < 8; ++i) {
    int m = m0 + half * 8 + i; if (m >= N) m = N - 1;
    float pb[5];
#pragma unroll
    for (int j = 0; j < 5; ++j) pb[j] = predBoxes[(size_t)m * 5 + j];
    float px[4], py[4];
    {
      const float4* pc = (const float4*)(predCorn + (size_t)m * 8);
      const float4 c0 = pc[0], c1 = pc[1];
      px[0] = c0.x; py[0] = c0.y; px[1] = c0.z; py[1] = c0.w;
      px[2] = c1.x; py[2] = c1.y; px[3] = c1.z; py[3] = c1.w;
    }
    const float pa = predArea[m];

    const float sp = fabsf(pb[0] - tb[0]) + fabsf(pb[1] - tb[1]) +
                     fabsf(pb[2] - tb[2]) + fabsf(pb[3] - tb[3]);
    float r = fmodf(pb[4] - tb[4] + HALF_PI_F, PI_F);
    if (r < 0.0f) r += PI_F;                       // python-style floor mod
    const float ca = fabsf(r - HALF_PI_F) * (2.0f / PI_F);

    const float inter = quadInterArea(px, py, qx, qy);
    const float iou = inter / (pa + ta - inter + 1e-8f);
    c[i] = 5.0f * (sp + ca) - 2.0f * iou;          // COST_BBOX, COST_GIOU
  }

  const int lbl = tgtLabels[t];
  bool usedWmma = false;
#if defined(__has_builtin) && __has_builtin(__builtin_amdgcn_wmma_f32_16x16x4_f32)
  if (C == 16) {
    // D = A x B + C with A[m,k] = ccS[m0+m, k] (16x16, K chained 4x4),
    // B[k,t] = onehot(label[t]==k). Exact f32 gather of the class cost.
    int mA = m0 + col; if (mA >= N) mA = N - 1;
    const float* arow = ccS + (size_t)mA * 16;
#pragma unroll
    for (int kk = 0; kk < 4; ++kk) {
      const int kbase = kk * 4 + half * 2;         // K split across lane halves
      v2f a, b;
      a.x = arow[kbase + 0];
      a.y = arow[kbase + 1];
      b.x = (lbl == kbase + 0) ? 1.0f : 0.0f;
      b.y = (lbl == kbase + 1) ? 1.0f : 0.0f;
      c = __builtin_amdgcn_wmma_f32_16x16x4_f32(
          /*neg_a=*/false, a, /*neg_b=*/false, b,
          /*c_mod=*/(short)0, c, /*reuse_a=*/false, /*reuse_b=*/false);
    }
    usedWmma = true;
  }
#endif
  if (!usedWmma) {
#pragma unroll
    for (int i = 0; i < 8; ++i) {
      int m = m0 + half * 8 + i; if (m >= N) m = N - 1;
      c[i] += ccS[(size_t)m * C + lbl];
    }
  }

#pragma unroll
  for (int i = 0; i < 8; ++i) {
    const int m = m0 + half * 8 + i;
    const int tc = t0 + col;
    if (m < N && tc < T) out[(size_t)m * T + tc] = c[i];
  }
}

extern "C" void kernel_launch(void* const* d_in, const int* in_sizes, int n_in,
                              void* d_out, int out_size, void* d_ws, size_t ws_size,
                              hipStream_t stream) {
  const float* logits    = (const float*)d_in[0];   // [8,300,16]
  const float* predBoxes = (const float*)d_in[1];   // [8,300,5]
  const int*   labels    = (const int*)d_in[2];     // [400]
  const float* tgtBoxes  = (const float*)d_in[3];   // [400,5]

  const int N = in_sizes[1] / 5;        // 2400
  const int C = in_sizes[0] / N;        // 16
  const int T = in_sizes[3] / 5;        // 400

  // Workspace layout (floats), offsets kept 16B-aligned for float4 loads.
  float* ws = (float*)d_ws;
  size_t off = 0;
  auto align4 = [](size_t v) { return (v + 3) & ~(size_t)3; };
  float* ccS      = ws + off; off = align4(off + (size_t)N * C);
  float* predCorn = ws + off; off = align4(off + (size_t)N * 8);
  float* predArea = ws + off; off = align4(off + (size_t)N);
  float* tgtCorn  = ws + off; off = align4(off + (size_t)T * 8);
  float* tgtArea  = ws + off; off = align4(off + (size_t)T);
  (void)ws_size;  // ~255 KB needed

  hm_prep_cc<<<(N * C + 255) / 256, 256, 0, stream>>>(logits, ccS, N * C);
  hm_prep_boxes<<<(N + T + 255) / 256, 256, 0, stream>>>(
      predBoxes, tgtBoxes, predCorn, predArea, tgtCorn, tgtArea, N, T);

  const int tilesM = (N + 15) / 16, tilesT = (T + 15) / 16;
  const int waves = tilesM * tilesT;                 // 3750 tiles
  const int wavesPerBlk = 4;                         // 128 threads
  const int blocks = (waves + wavesPerBlk - 1) / wavesPerBlk;
  hm_cost_tiles<<<blocks, 128, 0, stream>>>(
      predBoxes, tgtBoxes, labels, ccS, predCorn, predArea, tgtCorn, tgtArea,
      (float*)d_out, N, T, C, tilesT);
}